// CRFDecoderABC_66348654788679
// MI455X (gfx1250) — compile-verified
//
#include <hip/hip_runtime.h>
#include <math.h>

#define SLN 512
#define BSZ 128
#define TAG 32

typedef __attribute__((ext_vector_type(2))) float v2f;
typedef __attribute__((ext_vector_type(8))) float v8f;

// lane l <-> lane l^16 swap. On gfx1250 this is a single VALU
// v_permlanex16_b32 (identity lane-select, cross-row read) -- avoids the
// ds_bpermute + s_wait_dscnt round trip that __shfl_xor(.,16) generates.
__device__ __forceinline__ float swap16(float x) {
#if __has_builtin(__builtin_amdgcn_permlanex16)
    int xi = __float_as_int(x);
    int r  = __builtin_amdgcn_permlanex16(xi, xi, 0x76543210u, 0xFEDCBA98u,
                                          false, false);
    return __int_as_float(r);
#else
    return __shfl_xor(x, 16);
#endif
}

// Natural log for arguments known to be >= ~0.5 (never denormal): raw
// v_log_f32 (base-2) * ln2, skipping __logf's denormal-rescale preamble
// (cmp/cndmask/ldexp) that otherwise lands on the scan's critical path.
__device__ __forceinline__ float fast_ln(float x) {
#if __has_builtin(__builtin_amdgcn_logf)
    return __builtin_amdgcn_logf(x) * 0.6931471805599453f;
#else
    return __logf(x);
#endif
}

// ---------------------------------------------------------------------------
// Forward (alpha) scan in exp-space using V_WMMA_F32_16X16X4_F32.
// One wave (32 lanes) handles 16 batch rows. Layouts:
//   alpha / em / D-tile:  reg v, lane l :  tag = v + 8*(l>>4) (+16 for the
//                         second tile), batch = blockIdx.x*16 + (l&15)
//   A operand (const)  :  exp(transition)^T, M = out-tag, K = in-tag
//   B operand (per step): p = exp(alpha - rowmax), K = in-tag, N = batch
// D always contains the row-max term p=1, so D >= min(exp(T)) ~ 0.6 and the
// fast_ln path is numerically safe.
// ---------------------------------------------------------------------------
__global__ __launch_bounds__(32)
void crf_forward_kernel(const float* __restrict__ em,
                        const int*   __restrict__ len,
                        const float* __restrict__ T,
                        const float* __restrict__ startT,
                        const float* __restrict__ endT,
                        float*       __restrict__ lp_out)
{
    const int  lane = threadIdx.x;
    const int  half = lane >> 4;      // 0: tags v .. / 1: tags v+8 ..
    const int  bl   = lane & 15;      // batch-in-group == N column == A's M row
    const int  b    = blockIdx.x * 16 + bl;
    const bool lo   = (half == 0);

    // ---- loop-invariant A operands: A = exp(transition)^T --------------------
    // A0 -> out-tags 0..15, A1 -> out-tags 16..31. Chunk c covers K = 4c..4c+3.
    v2f A0[8], A1[8];
#pragma unroll
    for (int c = 0; c < 8; ++c) {
        const int k0 = 4 * c + 2 * half;
        A0[c][0] = __expf(T[(k0 + 0) * TAG + bl]);
        A0[c][1] = __expf(T[(k0 + 1) * TAG + bl]);
        A1[c][0] = __expf(T[(k0 + 0) * TAG + 16 + bl]);
        A1[c][1] = __expf(T[(k0 + 1) * TAG + 16 + bl]);
    }

    const float* eb   = em + (size_t)b * SLN * TAG;
    const int    lenb = len[b];

    // ---- alpha0 = start_transition + em[t=0] --------------------------------
    float al0[8], al1[8];
    {
        const float4* st0 = (const float4*)(startT + 8 * half);
        const float4* st1 = (const float4*)(startT + 16 + 8 * half);
        const float4* q0  = (const float4*)(eb + 8 * half);
        const float4* q1  = (const float4*)(eb + 16 + 8 * half);
        float4 sA = st0[0], sB = st0[1], sC = st1[0], sD = st1[1];
        float4 eA = q0[0],  eB = q0[1],  eC = q1[0],  eD = q1[1];
        al0[0]=sA.x+eA.x; al0[1]=sA.y+eA.y; al0[2]=sA.z+eA.z; al0[3]=sA.w+eA.w;
        al0[4]=sB.x+eB.x; al0[5]=sB.y+eB.y; al0[6]=sB.z+eB.z; al0[7]=sB.w+eB.w;
        al1[0]=sC.x+eC.x; al1[1]=sC.y+eC.y; al1[2]=sC.z+eC.z; al1[3]=sC.w+eC.w;
        al1[4]=sD.x+eD.x; al1[5]=sD.y+eD.y; al1[6]=sD.z+eD.z; al1[7]=sD.w+eD.w;
    }

    // ---- 511 serial scan steps ----------------------------------------------
    for (int t = 1; t < SLN; ++t) {
        // stream this step's emissions (and prefetch ahead to hide HBM latency)
        float em0[8], em1[8];
        {
            const float4* q0 = (const float4*)(eb + (size_t)t * TAG + 8 * half);
            const float4* q1 = (const float4*)(eb + (size_t)t * TAG + 16 + 8 * half);
            float4 eA = q0[0], eB = q0[1], eC = q1[0], eD = q1[1];
            em0[0]=eA.x; em0[1]=eA.y; em0[2]=eA.z; em0[3]=eA.w;
            em0[4]=eB.x; em0[5]=eB.y; em0[6]=eB.z; em0[7]=eB.w;
            em1[0]=eC.x; em1[1]=eC.y; em1[2]=eC.z; em1[3]=eC.w;
            em1[4]=eD.x; em1[5]=eD.y; em1[6]=eD.z; em1[7]=eD.w;
        }
        if (t + 8 < SLN)
            __builtin_prefetch(eb + (size_t)(t + 8) * TAG + 8 * half, 0, 1);

        // per-batch max: lane-local over 16 regs + one cross-half swap
        float mloc = al0[0];
#pragma unroll
        for (int v = 1; v < 8; ++v) mloc = fmaxf(mloc, al0[v]);
#pragma unroll
        for (int v = 0; v < 8; ++v) mloc = fmaxf(mloc, al1[v]);
        const float m = fmaxf(mloc, swap16(mloc));

        // p = exp(alpha - m)  (still in D-tile layout)
        float p0[8], p1[8];
#pragma unroll
        for (int v = 0; v < 8; ++v) { p0[v] = __expf(al0[v] - m); p1[v] = __expf(al1[v] - m); }

        // relayout p into the 8 B operand chunks (K = in-tag, N = batch)
        v2f Bc[8];
#pragma unroll
        for (int c = 0; c < 2; ++c) {            // k = 0..7   (source: p0, low half)
            const int v = 4 * c;
            float s0 = swap16(p0[v + 2]), s1 = swap16(p0[v + 3]);
            Bc[c][0] = lo ? p0[v]     : s0;
            Bc[c][1] = lo ? p0[v + 1] : s1;
        }
#pragma unroll
        for (int c = 2; c < 4; ++c) {            // k = 8..15  (source: p0, high half)
            const int v = 4 * (c - 2);
            float s0 = swap16(p0[v]), s1 = swap16(p0[v + 1]);
            Bc[c][0] = lo ? s0 : p0[v + 2];
            Bc[c][1] = lo ? s1 : p0[v + 3];
        }
#pragma unroll
        for (int c = 4; c < 6; ++c) {            // k = 16..23 (source: p1, low half)
            const int v = 4 * (c - 4);
            float s0 = swap16(p1[v + 2]), s1 = swap16(p1[v + 3]);
            Bc[c][0] = lo ? p1[v]     : s0;
            Bc[c][1] = lo ? p1[v + 1] : s1;
        }
#pragma unroll
        for (int c = 6; c < 8; ++c) {            // k = 24..31 (source: p1, high half)
            const int v = 4 * (c - 6);
            float s0 = swap16(p1[v]), s1 = swap16(p1[v + 1]);
            Bc[c][0] = lo ? s0 : p1[v + 2];
            Bc[c][1] = lo ? s1 : p1[v + 3];
        }

        // D = exp(T)^T x p : 2 output tiles, each as TWO independent 4-deep
        // WMMA chains (halves the dependent-WMMA latency on the scan's
        // critical path; the 4 chains pipeline through the matrix unit).
        v8f d0a, d0b, d1a, d1b;
#pragma unroll
        for (int i = 0; i < 8; ++i) { d0a[i] = 0.0f; d0b[i] = 0.0f; d1a[i] = 0.0f; d1b[i] = 0.0f; }
#pragma unroll
        for (int c = 0; c < 4; ++c) {
            d0a = __builtin_amdgcn_wmma_f32_16x16x4_f32(false, A0[c], false, Bc[c],
                                                        (short)0, d0a, false, false);
            d1a = __builtin_amdgcn_wmma_f32_16x16x4_f32(false, A1[c], false, Bc[c],
                                                        (short)0, d1a, false, false);
        }
#pragma unroll
        for (int c = 4; c < 8; ++c) {
            d0b = __builtin_amdgcn_wmma_f32_16x16x4_f32(false, A0[c], false, Bc[c],
                                                        (short)0, d0b, false, false);
            d1b = __builtin_amdgcn_wmma_f32_16x16x4_f32(false, A1[c], false, Bc[c],
                                                        (short)0, d1b, false, false);
        }

        // alpha' = m + log(D) + em  ; padded rows (t >= len) keep old alpha
        const bool upd = (t < lenb);
#pragma unroll
        for (int v = 0; v < 8; ++v) {
            float n0 = m + fast_ln(d0a[v] + d0b[v]) + em0[v];
            float n1 = m + fast_ln(d1a[v] + d1b[v]) + em1[v];
            al0[v] = upd ? n0 : al0[v];
            al1[v] = upd ? n1 : al1[v];
        }
    }

    // ---- log_partition = logsumexp(alpha + end_transition) ------------------
    {
        const float4* et0 = (const float4*)(endT + 8 * half);
        const float4* et1 = (const float4*)(endT + 16 + 8 * half);
        float4 eA = et0[0], eB = et0[1], eC = et1[0], eD = et1[1];
        float x0[8], x1[8];
        x0[0]=al0[0]+eA.x; x0[1]=al0[1]+eA.y; x0[2]=al0[2]+eA.z; x0[3]=al0[3]+eA.w;
        x0[4]=al0[4]+eB.x; x0[5]=al0[5]+eB.y; x0[6]=al0[6]+eB.z; x0[7]=al0[7]+eB.w;
        x1[0]=al1[0]+eC.x; x1[1]=al1[1]+eC.y; x1[2]=al1[2]+eC.z; x1[3]=al1[3]+eC.w;
        x1[4]=al1[4]+eD.x; x1[5]=al1[5]+eD.y; x1[6]=al1[6]+eD.z; x1[7]=al1[7]+eD.w;

        float mloc = x0[0];
#pragma unroll
        for (int v = 1; v < 8; ++v) mloc = fmaxf(mloc, x0[v]);
#pragma unroll
        for (int v = 0; v < 8; ++v) mloc = fmaxf(mloc, x1[v]);
        const float M = fmaxf(mloc, swap16(mloc));

        float s = 0.0f;
#pragma unroll
        for (int v = 0; v < 8; ++v) s += __expf(x0[v] - M) + __expf(x1[v] - M);
        s += swap16(s);

        if (lo) lp_out[b] = M + fast_ln(s);
    }
}

// ---------------------------------------------------------------------------
// Gold-path score: s + p + t + e  (one thread per batch, trivial gathers)
// ---------------------------------------------------------------------------
__global__ __launch_bounds__(128)
void crf_score_kernel(const float* __restrict__ em,
                      const int*   __restrict__ len,
                      const int*   __restrict__ tgt,
                      const float* __restrict__ T,
                      const float* __restrict__ startT,
                      const float* __restrict__ endT,
                      float*       __restrict__ score_out)
{
    const int b = blockIdx.x * blockDim.x + threadIdx.x;
    if (b >= BSZ) return;
    const int    L  = len[b];
    const int*   tg = tgt + (size_t)b * SLN;
    const float* e  = em + (size_t)b * SLN * TAG;

    int   prev = tg[0];
    float acc  = startT[prev] + e[prev];
    for (int t = 1; t < SLN; ++t) {
        if (t < L) {
            const int cur = tg[t];
            acc += e[(size_t)t * TAG + cur] + T[prev * TAG + cur];
            prev = cur;
        }
    }
    acc += endT[prev];
    score_out[b] = acc;
}

// ---------------------------------------------------------------------------
// Deterministic fixed-order reduction of sum_b (log_partition - log_score)
// ---------------------------------------------------------------------------
__global__ __launch_bounds__(128)
void crf_reduce_kernel(const float* __restrict__ lp,
                       const float* __restrict__ sc,
                       float*       __restrict__ out)
{
    __shared__ float buf[BSZ];
    const int i = threadIdx.x;
    buf[i] = lp[i] - sc[i];
    __syncthreads();
    for (int s = BSZ / 2; s > 0; s >>= 1) {
        if (i < s) buf[i] += buf[i + s];
        __syncthreads();
    }
    if (i == 0) out[0] = buf[0];
}

extern "C" void kernel_launch(void* const* d_in, const int* in_sizes, int n_in,
                              void* d_out, int out_size, void* d_ws, size_t ws_size,
                              hipStream_t stream) {
    const float* emission = (const float*)d_in[0];
    const int*   length   = (const int*)  d_in[1];
    const int*   target   = (const int*)  d_in[2];
    const float* trans    = (const float*)d_in[3];
    const float* startT   = (const float*)d_in[4];
    const float* endT     = (const float*)d_in[5];

    float* lp    = (float*)d_ws;        // BSZ floats: log partition per batch
    float* score = lp + BSZ;            // BSZ floats: gold-path score per batch

    crf_forward_kernel<<<BSZ / 16, 32, 0, stream>>>(emission, length, trans,
                                                    startT, endT, lp);
    crf_score_kernel<<<1, 128, 0, stream>>>(emission, length, target, trans,
                                            startT, endT, score);
    crf_reduce_kernel<<<1, 128, 0, stream>>>(lp, score, (float*)d_out);
}